// B_Conv2d_ConvNN_K_N_20435454394603
// MI455X (gfx1250) — compile-verified
//
#include <hip/hip_runtime.h>

typedef float v2f __attribute__((ext_vector_type(2)));
typedef float v8f __attribute__((ext_vector_type(8)));

// D = A(16x4) * B(4x16) + C, fp32 WMMA (CDNA5 V_WMMA_F32_16X16X4_F32)
__device__ __forceinline__ v8f wmma4f32(v2f a, v2f b, v8f c) {
  return __builtin_amdgcn_wmma_f32_16x16x4_f32(false, a, false, b, (short)0, c,
                                               false, false);
}

// ---------------------------------------------------------------------------
// K1: pixel_unshuffle(x,2) -> X1 (B,12,256) and sample extraction S1 (B,12,64)
// ---------------------------------------------------------------------------
__global__ __launch_bounds__(256) void k_unshuffle_samp1(
    const float* __restrict__ x, const int* __restrict__ idx1,
    float* __restrict__ X1, float* __restrict__ S1) {
  int b = blockIdx.x;
  const float* xb = x + (size_t)b * 3 * 32 * 32;
  float* X1b = X1 + (size_t)b * 3072;
  for (int e = threadIdx.x; e < 3072; e += 256) {
    int ch = e >> 8, t = e & 255;
    int ho = t >> 4, wo = t & 15;
    int c = ch >> 2, rh = (ch >> 1) & 1, rw = ch & 1;
    X1b[e] = xb[c * 1024 + (2 * ho + rh) * 32 + (2 * wo + rw)];
  }
  float* S1b = S1 + (size_t)b * 768;
  for (int e = threadIdx.x; e < 768; e += 256) {
    int ch = e >> 6, n = e & 63;
    int t = idx1[n];
    int ho = t >> 4, wo = t & 15;
    int c = ch >> 2, rh = (ch >> 1) & 1, rw = ch & 1;
    S1b[e] = xb[c * 1024 + (2 * ho + rh) * 32 + (2 * wo + rw)];
  }
}

// ---------------------------------------------------------------------------
// K2: layer1 3x3 conv as implicit GEMM, M=16 out-ch, K=108, N=256 pixels.
//     Padded 18x18 LDS image -> branchless, division-free inner loop.
//     K order: (taps 0..7 per ch) then (tap 8 for 4 ch per step): 27 wmma/tile
// ---------------------------------------------------------------------------
__global__ __launch_bounds__(256) void k_conv3x3_l1(
    const float* __restrict__ X1, const float* __restrict__ w1a,
    const float* __restrict__ b1a, float* __restrict__ Y1) {
  __shared__ float img[12 * 324];  // 12 ch x 18x18 padded (15.2KB)
  __shared__ float wA[1728];       // 16 x 108
  int b = blockIdx.x;
  const float* X1b = X1 + (size_t)b * 3072;
  for (int e = threadIdx.x; e < 3888; e += 256) img[e] = 0.f;
  for (int e = threadIdx.x; e < 1728; e += 256) wA[e] = w1a[e];
  __syncthreads();
  for (int e = threadIdx.x; e < 3072; e += 256) {
    int c = e >> 8, t = e & 255;
    img[c * 324 + ((t >> 4) + 1) * 18 + (t & 15) + 1] = X1b[e];
  }
  __syncthreads();
  int lane = threadIdx.x & 31, wv = threadIdx.x >> 5;
  int hlf = lane >> 4, mn = lane & 15;
  // per-lane tap->padded-offset constants (hoisted selects)
  int d00 = hlf ? 2 : 0, d01 = hlf ? 18 : 1;    // taps {0,1}|{2,3}
  int d10 = hlf ? 36 : 19, d11 = hlf ? 37 : 20; // taps {4,5}|{6,7}
  int tb0 = 2 * hlf;
  for (int tile = wv * 2; tile < wv * 2 + 2; ++tile) {
    int t0 = tile * 16;
    int t = t0 + mn;
    int pb = (t >> 4) * 18 + (t & 15);
    v8f acc = {};
    const float* Arow = wA + mn * 108;
    for (int c = 0; c < 12; ++c) {  // taps 0..7
      const float* ic = img + c * 324 + pb;
      int a0 = c * 9 + tb0;
      v2f a, bf;
      a.x = Arow[a0];     a.y = Arow[a0 + 1];
      bf.x = ic[d00];     bf.y = ic[d01];
      acc = wmma4f32(a, bf, acc);
      a.x = Arow[a0 + 4]; a.y = Arow[a0 + 5];
      bf.x = ic[d10];     bf.y = ic[d11];
      acc = wmma4f32(a, bf, acc);
    }
    for (int q = 0; q < 3; ++q) {   // tap 8, 4 channels per step
      int c0 = q * 4 + 2 * hlf;
      v2f a, bf;
      a.x = Arow[c0 * 9 + 8];
      a.y = Arow[c0 * 9 + 17];
      bf.x = img[c0 * 324 + pb + 38];
      bf.y = img[c0 * 324 + pb + 362];
      acc = wmma4f32(a, bf, acc);
    }
    float* out = Y1 + (size_t)b * 8192;
#pragma unroll
    for (int v = 0; v < 8; ++v) {
      int o = v + 8 * hlf;
      float r = acc[v] + b1a[o];
      out[o * 256 + t0 + mn] = r > 0.f ? r : 0.f;
    }
  }
}

// ---------------------------------------------------------------------------
// K3: layer1 KNN: per (b,t) squared distances to 64 samples, top-9 indices
// ---------------------------------------------------------------------------
__global__ __launch_bounds__(256) void k_knn_l1(const float* __restrict__ X1,
                                                const float* __restrict__ S1,
                                                int* __restrict__ NBR1) {
  __shared__ float sp[768];
  int b = blockIdx.x;
  for (int e = threadIdx.x; e < 768; e += 256) sp[e] = S1[(size_t)b * 768 + e];
  __syncthreads();
  int t = threadIdx.x;
  float xv[12];
#pragma unroll
  for (int c = 0; c < 12; ++c) xv[c] = X1[(size_t)b * 3072 + c * 256 + t];
  float bd[9];
  int bi[9];
#pragma unroll
  for (int k = 0; k < 9; ++k) { bd[k] = 3.4e38f; bi[k] = 0; }
  for (int j = 0; j < 64; ++j) {
    float d = 0.f;
#pragma unroll
    for (int c = 0; c < 12; ++c) {
      float df = xv[c] - sp[c * 64 + j];
      d = fmaf(df, df, d);
    }
    float cd = d; int ci = j;
#pragma unroll
    for (int k = 0; k < 9; ++k) {
      if (cd < bd[k]) {
        float td = bd[k]; int ti = bi[k];
        bd[k] = cd; bi[k] = ci; cd = td; ci = ti;
      }
    }
  }
  int* o = NBR1 + ((size_t)b * 256 + t) * 9;
#pragma unroll
  for (int k = 0; k < 9; ++k) o[k] = bi[k];
}

// ---------------------------------------------------------------------------
// K4: layer1 gather-GEMM: y2 = w1b(16x108) @ G(108x256), relu -> Y1 [16,32)
//     Neighbor indices in registers; hlf-selects hoisted; division-free.
// ---------------------------------------------------------------------------
__global__ __launch_bounds__(256) void k_knn_gemm_l1(
    const float* __restrict__ S1, const int* __restrict__ NBR1,
    const float* __restrict__ w1b, const float* __restrict__ b1b,
    float* __restrict__ Y1) {
  __shared__ float sp[768];
  __shared__ float wB[1728];
  __shared__ int nbL[2304];
  int b = blockIdx.x;
  for (int e = threadIdx.x; e < 768; e += 256) sp[e] = S1[(size_t)b * 768 + e];
  for (int e = threadIdx.x; e < 1728; e += 256) wB[e] = w1b[e];
  for (int e = threadIdx.x; e < 2304; e += 256)
    nbL[e] = NBR1[(size_t)b * 2304 + e];
  __syncthreads();
  int lane = threadIdx.x & 31, wv = threadIdx.x >> 5;
  int hlf = lane >> 4, mn = lane & 15;
  int tb0 = 2 * hlf;
  for (int tile = wv * 2; tile < wv * 2 + 2; ++tile) {
    int t0 = tile * 16;
    int t = t0 + mn;
    const int* nbp = nbL + t * 9;
    int nb0 = nbp[0], nb1 = nbp[1], nb2 = nbp[2], nb3 = nbp[3];
    int nb4 = nbp[4], nb5 = nbp[5], nb6 = nbp[6], nb7 = nbp[7], nb8 = nbp[8];
    int j00 = hlf ? nb2 : nb0, j01 = hlf ? nb3 : nb1;
    int j10 = hlf ? nb6 : nb4, j11 = hlf ? nb7 : nb5;
    v8f acc = {};
    const float* Arow = wB + mn * 108;
    for (int c = 0; c < 12; ++c) {  // taps 0..7
      int cb = c * 64;
      int a0 = c * 9 + tb0;
      v2f a, bf;
      a.x = Arow[a0];     a.y = Arow[a0 + 1];
      bf.x = sp[cb + j00]; bf.y = sp[cb + j01];
      acc = wmma4f32(a, bf, acc);
      a.x = Arow[a0 + 4]; a.y = Arow[a0 + 5];
      bf.x = sp[cb + j10]; bf.y = sp[cb + j11];
      acc = wmma4f32(a, bf, acc);
    }
    for (int q = 0; q < 3; ++q) {   // tap 8, 4 channels per step
      int c0 = q * 4 + 2 * hlf;
      v2f a, bf;
      a.x = Arow[c0 * 9 + 8];
      a.y = Arow[c0 * 9 + 17];
      bf.x = sp[c0 * 64 + nb8];
      bf.y = sp[c0 * 64 + 64 + nb8];
      acc = wmma4f32(a, bf, acc);
    }
    float* out = Y1 + (size_t)b * 8192;
#pragma unroll
    for (int v = 0; v < 8; ++v) {
      int o = v + 8 * hlf;
      float r = acc[v] + b1b[o];
      out[(16 + o) * 256 + t0 + mn] = r > 0.f ? r : 0.f;
    }
  }
}

// ---------------------------------------------------------------------------
// K5: layer1 1x1 conv: Z1 = w1p(64x32) @ Y1(32x256) + bias  (no relu)
//     (pixel_shuffle then unshuffle is identity -> Z1 is layer2's input)
// ---------------------------------------------------------------------------
__global__ __launch_bounds__(256) void k_conv1x1_l1(
    const float* __restrict__ Y1, const float* __restrict__ w1p,
    const float* __restrict__ b1p, float* __restrict__ Z1) {
  __shared__ float ybuf[8192];  // 32 x 256
  __shared__ float wP[2048];    // 64 x 32
  int b = blockIdx.x;
  for (int e = threadIdx.x; e < 8192; e += 256) ybuf[e] = Y1[(size_t)b * 8192 + e];
  for (int e = threadIdx.x; e < 2048; e += 256) wP[e] = w1p[e];
  __syncthreads();
  int lane = threadIdx.x & 31, wv = threadIdx.x >> 5;
  int hlf = lane >> 4, mn = lane & 15;
  for (int tl = wv * 8; tl < wv * 8 + 8; ++tl) {
    int mt = tl >> 4, nt = tl & 15;
    int t0 = nt * 16;
    v8f acc = {};
#pragma unroll
    for (int kk = 0; kk < 32; kk += 4) {
      int k0 = kk + 2 * hlf;
      v2f a, bf;
      a.x = wP[(mt * 16 + mn) * 32 + k0];
      a.y = wP[(mt * 16 + mn) * 32 + k0 + 1];
      bf.x = ybuf[k0 * 256 + t0 + mn];
      bf.y = ybuf[(k0 + 1) * 256 + t0 + mn];
      acc = wmma4f32(a, bf, acc);
    }
    float* out = Z1 + (size_t)b * 16384;
#pragma unroll
    for (int v = 0; v < 8; ++v) {
      int o = mt * 16 + v + 8 * hlf;
      out[o * 256 + t0 + mn] = acc[v] + b1p[o];
    }
  }
}

// ---------------------------------------------------------------------------
// K6: layer2 sample extraction S2 (B,64,64)
// ---------------------------------------------------------------------------
__global__ __launch_bounds__(256) void k_samp2(const float* __restrict__ Z1,
                                               const int* __restrict__ idx2,
                                               float* __restrict__ S2) {
  int b = blockIdx.x;
  for (int e = threadIdx.x; e < 4096; e += 256) {
    int c = e >> 6, n = e & 63;
    S2[(size_t)b * 4096 + e] = Z1[(size_t)b * 16384 + c * 256 + idx2[n]];
  }
}

// ---------------------------------------------------------------------------
// K7: layer2 3x3 conv, M=32 out-ch, K=576 -> 144 wmma/tile, zero waste.
//     Two 32-channel phases of padded LDS image (40.5KB); acc kept in VGPRs.
// ---------------------------------------------------------------------------
__global__ __launch_bounds__(256) void k_conv3x3_l2(
    const float* __restrict__ Z1, const float* __restrict__ w2a,
    const float* __restrict__ b2a, float* __restrict__ Y2) {
  __shared__ float img[32 * 324];  // 32 ch x 18x18 padded = 40.5KB
  int b = blockIdx.x;
  int lane = threadIdx.x & 31, wv = threadIdx.x >> 5;
  int hlf = lane >> 4, mn = lane & 15;
  int d00 = hlf ? 2 : 0, d01 = hlf ? 18 : 1;
  int d10 = hlf ? 36 : 19, d11 = hlf ? 37 : 20;
  int tb0 = 2 * hlf;
  v8f acc[4];
  v8f zz = {};
#pragma unroll
  for (int i = 0; i < 4; ++i) acc[i] = zz;
  for (int e = threadIdx.x; e < 10368; e += 256) img[e] = 0.f;  // halo once
  for (int p = 0; p < 2; ++p) {
    __syncthreads();
    const float* src = Z1 + (size_t)b * 16384 + p * 8192;
    for (int e = threadIdx.x; e < 8192; e += 256) {
      int c = e >> 8, t = e & 255;
      img[c * 324 + ((t >> 4) + 1) * 18 + (t & 15) + 1] = src[e];
    }
    __syncthreads();
#pragma unroll
    for (int tli = 0; tli < 4; ++tli) {
      int tl = wv * 4 + tli;
      int mt = tl >> 4, nt = tl & 15;
      int t = nt * 16 + mn;
      int pb = (t >> 4) * 18 + (t & 15);
      const float* Arow = w2a + (size_t)(mt * 16 + mn) * 576 + p * 288;
      for (int c = 0; c < 32; ++c) {  // taps 0..7
        const float* ic = img + c * 324 + pb;
        int a0 = c * 9 + tb0;
        v2f a, bf;
        a.x = Arow[a0];     a.y = Arow[a0 + 1];
        bf.x = ic[d00];     bf.y = ic[d01];
        acc[tli] = wmma4f32(a, bf, acc[tli]);
        a.x = Arow[a0 + 4]; a.y = Arow[a0 + 5];
        bf.x = ic[d10];     bf.y = ic[d11];
        acc[tli] = wmma4f32(a, bf, acc[tli]);
      }
      for (int q = 0; q < 8; ++q) {  // tap 8, 4 channels per step
        int c0 = q * 4 + 2 * hlf;
        v2f a, bf;
        a.x = Arow[c0 * 9 + 8];
        a.y = Arow[c0 * 9 + 17];
        bf.x = img[c0 * 324 + pb + 38];
        bf.y = img[c0 * 324 + pb + 362];
        acc[tli] = wmma4f32(a, bf, acc[tli]);
      }
    }
  }
  float* out = Y2 + (size_t)b * 16384;
#pragma unroll
  for (int tli = 0; tli < 4; ++tli) {
    int tl = wv * 4 + tli;
    int mt = tl >> 4, nt = tl & 15;
#pragma unroll
    for (int v = 0; v < 8; ++v) {
      int o = mt * 16 + v + 8 * hlf;
      float r = acc[tli][v] + b2a[o];
      out[o * 256 + nt * 16 + mn] = r > 0.f ? r : 0.f;
    }
  }
}

// ---------------------------------------------------------------------------
// K8: layer2 KNN top-9
// ---------------------------------------------------------------------------
__global__ __launch_bounds__(256) void k_knn_l2(const float* __restrict__ Z1,
                                                const float* __restrict__ S2,
                                                int* __restrict__ NBR2) {
  __shared__ float sp[4096];  // 16KB
  int b = blockIdx.x;
  for (int e = threadIdx.x; e < 4096; e += 256) sp[e] = S2[(size_t)b * 4096 + e];
  __syncthreads();
  int t = threadIdx.x;
  float xv[64];
#pragma unroll
  for (int c = 0; c < 64; ++c) xv[c] = Z1[(size_t)b * 16384 + c * 256 + t];
  float bd[9];
  int bi[9];
#pragma unroll
  for (int k = 0; k < 9; ++k) { bd[k] = 3.4e38f; bi[k] = 0; }
  for (int j = 0; j < 64; ++j) {
    float d = 0.f;
#pragma unroll
    for (int c = 0; c < 64; ++c) {
      float df = xv[c] - sp[c * 64 + j];
      d = fmaf(df, df, d);
    }
    float cd = d; int ci = j;
#pragma unroll
    for (int k = 0; k < 9; ++k) {
      if (cd < bd[k]) {
        float td = bd[k]; int ti = bi[k];
        bd[k] = cd; bi[k] = ci; cd = td; ci = ti;
      }
    }
  }
  int* o = NBR2 + ((size_t)b * 256 + t) * 9;
#pragma unroll
  for (int k = 0; k < 9; ++k) o[k] = bi[k];
}

// ---------------------------------------------------------------------------
// K9: layer2 gather-GEMM: w2b(32x576) @ G(576x256), relu -> Y2 ch [32,64)
// ---------------------------------------------------------------------------
__global__ __launch_bounds__(256) void k_knn_gemm_l2(
    const float* __restrict__ S2, const int* __restrict__ NBR2,
    const float* __restrict__ w2b, const float* __restrict__ b2b,
    float* __restrict__ Y2) {
  __shared__ float sp[4096];  // 16KB
  __shared__ int nbL[2304];   // 9KB
  int b = blockIdx.x;
  for (int e = threadIdx.x; e < 4096; e += 256) sp[e] = S2[(size_t)b * 4096 + e];
  for (int e = threadIdx.x; e < 2304; e += 256)
    nbL[e] = NBR2[(size_t)b * 2304 + e];
  __syncthreads();
  int lane = threadIdx.x & 31, wv = threadIdx.x >> 5;
  int hlf = lane >> 4, mn = lane & 15;
  int tb0 = 2 * hlf;
  for (int tli = 0; tli < 4; ++tli) {
    int tl = wv * 4 + tli;
    int mt = tl >> 4, nt = tl & 15;
    int t0 = nt * 16;
    int t = t0 + mn;
    const int* nbp = nbL + t * 9;
    int nb0 = nbp[0], nb1 = nbp[1], nb2 = nbp[2], nb3 = nbp[3];
    int nb4 = nbp[4], nb5 = nbp[5], nb6 = nbp[6], nb7 = nbp[7], nb8 = nbp[8];
    int j00 = hlf ? nb2 : nb0, j01 = hlf ? nb3 : nb1;
    int j10 = hlf ? nb6 : nb4, j11 = hlf ? nb7 : nb5;
    v8f acc = {};
    const float* Arow = w2b + (size_t)(mt * 16 + mn) * 576;
    for (int c = 0; c < 64; ++c) {  // taps 0..7
      int cb = c * 64;
      int a0 = c * 9 + tb0;
      v2f a, bf;
      a.x = Arow[a0];      a.y = Arow[a0 + 1];
      bf.x = sp[cb + j00]; bf.y = sp[cb + j01];
      acc = wmma4f32(a, bf, acc);
      a.x = Arow[a0 + 4];  a.y = Arow[a0 + 5];
      bf.x = sp[cb + j10]; bf.y = sp[cb + j11];
      acc = wmma4f32(a, bf, acc);
    }
    for (int q = 0; q < 16; ++q) {  // tap 8, 4 channels per step
      int c0 = q * 4 + 2 * hlf;
      v2f a, bf;
      a.x = Arow[c0 * 9 + 8];
      a.y = Arow[c0 * 9 + 17];
      bf.x = sp[c0 * 64 + nb8];
      bf.y = sp[c0 * 64 + 64 + nb8];
      acc = wmma4f32(a, bf, acc);
    }
    float* out = Y2 + (size_t)b * 16384;
#pragma unroll
    for (int v = 0; v < 8; ++v) {
      int o = mt * 16 + v + 8 * hlf;
      float r = acc[v] + b2b[o];
      out[(32 + o) * 256 + t0 + mn] = r > 0.f ? r : 0.f;
    }
  }
}

// ---------------------------------------------------------------------------
// K10: layer2 1x1 conv: w2p(128x64) @ Y2(64x256) + bias, stored with the
//      final pixel_shuffle folded in and K-major transposed: Ht[k, b]
// ---------------------------------------------------------------------------
__global__ __launch_bounds__(256) void k_conv1x1_l2(
    const float* __restrict__ Y2, const float* __restrict__ w2p,
    const float* __restrict__ b2p, float* __restrict__ Ht) {
  __shared__ float ybuf[16384];  // 64KB
  int b = blockIdx.x;
  for (int e = threadIdx.x; e < 16384; e += 256) ybuf[e] = Y2[(size_t)b * 16384 + e];
  __syncthreads();
  int lane = threadIdx.x & 31, wv = threadIdx.x >> 5;
  int hlf = lane >> 4, mn = lane & 15;
  for (int tl = wv * 16; tl < wv * 16 + 16; ++tl) {
    int mt = tl >> 4, nt = tl & 15;
    int t0 = nt * 16;
    v8f acc = {};
    const float* Arow = w2p + (size_t)(mt * 16 + mn) * 64;
#pragma unroll
    for (int kk = 0; kk < 64; kk += 4) {
      int k0 = kk + 2 * hlf;
      v2f a, bf;
      a.x = Arow[k0];
      a.y = Arow[k0 + 1];
      bf.x = ybuf[k0 * 256 + t0 + mn];
      bf.y = ybuf[(k0 + 1) * 256 + t0 + mn];
      acc = wmma4f32(a, bf, acc);
    }
    int t = t0 + mn, ho = t >> 4, wo = t & 15;
#pragma unroll
    for (int v = 0; v < 8; ++v) {
      int ch = mt * 16 + v + 8 * hlf;
      int c2 = ch >> 2, rh = (ch >> 1) & 1, rw = ch & 1;
      int hidx = c2 * 1024 + (2 * ho + rh) * 32 + (2 * wo + rw);
      Ht[(size_t)hidx * 256 + b] = acc[v] + b2p[ch];
    }
  }
}

// ---------------------------------------------------------------------------
// K11: FC1 split-K GEMM. Out(1024 j x 256 b) = fc1w(1024x32768) @ Ht(32768x256)
//      grid (16 j-groups of 64, 8 k-slices of 4096), 16 waves per block.
//      Weight chunks staged in LDS; per-slice partials (deterministic).
// ---------------------------------------------------------------------------
__global__ __launch_bounds__(512) void k_fc1(const float* __restrict__ Ht,
                                             const float* __restrict__ fc1w,
                                             float* __restrict__ F1p) {
  __shared__ float wL[8192];  // 64 j x 128 k = 32KB
  int j0 = blockIdx.x * 64;
  int ks = blockIdx.y;
  int lane = threadIdx.x & 31, wv = threadIdx.x >> 5;
  int hlf = lane >> 4, mn = lane & 15;
  int b0 = wv * 16;
  v8f acc0 = {}, acc1 = {}, acc2 = {}, acc3 = {};
  for (int chk = 0; chk < 32; ++chk) {
    int kbase = ks * 4096 + chk * 128;
    __syncthreads();
    for (int e = threadIdx.x; e < 8192; e += 512) {
      int jj = e >> 7, kk = e & 127;
      wL[e] = fc1w[(size_t)(j0 + jj) * 32768 + kbase + kk];
    }
    if (chk + 1 < 32) {  // prefetch next weight chunk into cache
      int e = threadIdx.x * 16;
      int jj = e >> 7, kk = e & 127;
      __builtin_prefetch(&fc1w[(size_t)(j0 + jj) * 32768 + kbase + 128 + kk], 0, 1);
    }
    __syncthreads();
    for (int s = 0; s < 32; ++s) {
      int k0 = s * 4 + 2 * hlf;
      v2f bf, a;
      bf.x = Ht[(size_t)(kbase + k0) * 256 + b0 + mn];
      bf.y = Ht[(size_t)(kbase + k0 + 1) * 256 + b0 + mn];
      a.x = wL[mn * 128 + k0];          a.y = wL[mn * 128 + k0 + 1];
      acc0 = wmma4f32(a, bf, acc0);
      a.x = wL[(16 + mn) * 128 + k0];   a.y = wL[(16 + mn) * 128 + k0 + 1];
      acc1 = wmma4f32(a, bf, acc1);
      a.x = wL[(32 + mn) * 128 + k0];   a.y = wL[(32 + mn) * 128 + k0 + 1];
      acc2 = wmma4f32(a, bf, acc2);
      a.x = wL[(48 + mn) * 128 + k0];   a.y = wL[(48 + mn) * 128 + k0 + 1];
      acc3 = wmma4f32(a, bf, acc3);
    }
  }
  float* outp = F1p + (size_t)ks * 262144;
#pragma unroll
  for (int v = 0; v < 8; ++v) {
    int m = v + 8 * hlf;
    size_t row = (size_t)(b0 + mn) * 1024 + j0;
    outp[row + m]      = acc0[v];
    outp[row + 16 + m] = acc1[v];
    outp[row + 32 + m] = acc2[v];
    outp[row + 48 + m] = acc3[v];
  }
}

// ---------------------------------------------------------------------------
// K12: reduce the 8 split-K partials in fixed order, add bias, relu -> F1
// ---------------------------------------------------------------------------
__global__ __launch_bounds__(256) void k_reduce_f1(
    const float* __restrict__ F1p, const float* __restrict__ fc1b,
    float* __restrict__ F1) {
  int i = blockIdx.x * 256 + threadIdx.x;  // 262144
  float s = fc1b[i & 1023];
#pragma unroll
  for (int ks = 0; ks < 8; ++ks) s += F1p[(size_t)ks * 262144 + i];
  F1[i] = s > 0.f ? s : 0.f;
}

// ---------------------------------------------------------------------------
// K13: FC2 (tiny): out(256x10) = relu'd F1 @ fc2w^T + fc2b
// ---------------------------------------------------------------------------
__global__ __launch_bounds__(256) void k_fc2(const float* __restrict__ F1,
                                             const float* __restrict__ fc2w,
                                             const float* __restrict__ fc2b,
                                             float* __restrict__ out) {
  int id = blockIdx.x * 256 + threadIdx.x;
  if (id >= 2560) return;
  int b = id / 10, j = id % 10;
  const float* f = F1 + (size_t)b * 1024;
  const float* w = fc2w + (size_t)j * 1024;
  float s = fc2b[j];
  for (int k = 0; k < 1024; ++k) s = fmaf(f[k], w[k], s);
  out[id] = s;
}

// ---------------------------------------------------------------------------
extern "C" void kernel_launch(void* const* d_in, const int* in_sizes, int n_in,
                              void* d_out, int out_size, void* d_ws,
                              size_t ws_size, hipStream_t stream) {
  (void)in_sizes; (void)n_in; (void)out_size; (void)ws_size;
  const float* x    = (const float*)d_in[0];
  const int*   idx1 = (const int*)d_in[1];
  const int*   idx2 = (const int*)d_in[2];
  const float* w1a  = (const float*)d_in[3];
  const float* b1a  = (const float*)d_in[4];
  const float* w1b  = (const float*)d_in[5];
  const float* b1b  = (const float*)d_in[6];
  const float* w1p  = (const float*)d_in[7];
  const float* b1p  = (const float*)d_in[8];
  const float* w2a  = (const float*)d_in[9];
  const float* b2a  = (const float*)d_in[10];
  const float* w2b  = (const float*)d_in[11];
  const float* b2b  = (const float*)d_in[12];
  const float* w2p  = (const float*)d_in[13];
  const float* b2p  = (const float*)d_in[14];
  const float* fc1w = (const float*)d_in[15];
  const float* fc1b = (const float*)d_in[16];
  const float* fc2w = (const float*)d_in[17];
  const float* fc2b = (const float*)d_in[18];

  float* ws   = (float*)d_ws;
  float* X1   = ws;                        // 256*12*256   =   786432
  float* S1   = X1 + 786432;               // 256*12*64    =   196608
  int*   NBR1 = (int*)(S1 + 196608);       // 256*256*9    =   589824
  float* Y1   = (float*)(NBR1 + 589824);   // 256*32*256   =  2097152
  float* Z1   = Y1 + 2097152;              // 256*64*256   =  4194304
  float* S2   = Z1 + 4194304;              // 256*64*64    =  1048576
  int*   NBR2 = (int*)(S2 + 1048576);      // 589824
  float* Y2   = (float*)(NBR2 + 589824);   // 256*64*256   =  4194304
  float* Ht   = Y2 + 4194304;              // 32768*256    =  8388608
  float* F1p  = Ht + 8388608;              // 8*256*1024   =  2097152
  float* F1   = F1p + 2097152;             // 256*1024     =   262144

  k_unshuffle_samp1<<<256, 256, 0, stream>>>(x, idx1, X1, S1);
  k_conv3x3_l1<<<256, 256, 0, stream>>>(X1, w1a, b1a, Y1);
  k_knn_l1<<<256, 256, 0, stream>>>(X1, S1, NBR1);
  k_knn_gemm_l1<<<256, 256, 0, stream>>>(S1, NBR1, w1b, b1b, Y1);
  k_conv1x1_l1<<<256, 256, 0, stream>>>(Y1, w1p, b1p, Z1);
  k_samp2<<<256, 256, 0, stream>>>(Z1, idx2, S2);
  k_conv3x3_l2<<<256, 256, 0, stream>>>(Z1, w2a, b2a, Y2);
  k_knn_l2<<<256, 256, 0, stream>>>(Z1, S2, NBR2);
  k_knn_gemm_l2<<<256, 256, 0, stream>>>(S2, NBR2, w2b, b2b, Y2);
  k_conv1x1_l2<<<256, 256, 0, stream>>>(Y2, w2p, b2p, Ht);
  k_fc1<<<dim3(16, 8), 512, 0, stream>>>(Ht, fc1w, F1p);
  k_reduce_f1<<<1024, 256, 0, stream>>>(F1p, fc1b, F1);
  k_fc2<<<10, 256, 0, stream>>>(F1, fc2w, fc2b, (float*)d_out);
}